// SparseMoE_24859270710000
// MI455X (gfx1250) — compile-verified
//
#include <hip/hip_runtime.h>
#include <hip/hip_bf16.h>

typedef __attribute__((ext_vector_type(2))) float v2f;
typedef __attribute__((ext_vector_type(8))) float v8f;

#define B_   4
#define T_   1024
#define C_   768
#define H_   3072
#define E_   8
#define NT_  (B_ * T_)        // 4096 tokens
#define MAXA_ (NT_ * 2)       // worst-case assignments per expert
#define TM_  32               // tokens per tile (M)
#define HC_  128              // H-chunk for fused GEMM1->GEMM2
#define XSS_ (C_ + 4)         // 772: LDS stride for x tile (conflict-free)
#define HSS_ (HC_ + 4)        // 132: LDS stride for h chunk (conflict-free)

static __device__ __forceinline__ v8f wmma_f32_k4(v2f a, v2f b, v8f c) {
  // D(16x16,f32) = A(16x4,f32) * B(4x16,f32) + C
  return __builtin_amdgcn_wmma_f32_16x16x4_f32(false, a, false, b, (short)0, c,
                                               false, false);
}

// ---------------------------------------------------------------------------
// Kernel 1: router. One wave32 per token: logits = x[t] . Wg, softmax, top-2.
// Also zeroes the per-expert atomic counters.
// ---------------------------------------------------------------------------
__global__ __launch_bounds__(256) void moe_router_kernel(
    const float* __restrict__ x, const float* __restrict__ Wg,
    float* __restrict__ gates, int* __restrict__ eids, int* __restrict__ cnt) {
  const int gid  = blockIdx.x * blockDim.x + threadIdx.x;
  if (gid < E_) cnt[gid] = 0;
  const int t    = gid >> 5;
  const int lane = threadIdx.x & 31;
  if (t >= NT_) return;

  float acc[E_];
#pragma unroll
  for (int e = 0; e < E_; ++e) acc[e] = 0.0f;

  const float* xr = x + (size_t)t * C_;
  for (int c = lane; c < C_; c += 32) {
    const float xv = xr[c];
#pragma unroll
    for (int e = 0; e < E_; ++e) acc[e] += xv * Wg[c * E_ + e];
  }
#pragma unroll
  for (int e = 0; e < E_; ++e) {
    for (int off = 16; off > 0; off >>= 1)
      acc[e] += __shfl_xor(acc[e], off, 32);
  }
  if (lane == 0) {
    float m = acc[0];
#pragma unroll
    for (int e = 1; e < E_; ++e) m = fmaxf(m, acc[e]);
    float p[E_], s = 0.0f;
#pragma unroll
    for (int e = 0; e < E_; ++e) { p[e] = expf(acc[e] - m); s += p[e]; }
    const float inv = 1.0f / s;
    int i0 = 0; float v0 = p[0];
#pragma unroll
    for (int e = 1; e < E_; ++e) if (p[e] > v0) { v0 = p[e]; i0 = e; }
    int i1 = (i0 == 0) ? 1 : 0; float v1 = p[i1];
#pragma unroll
    for (int e = 0; e < E_; ++e)
      if (e != i0 && p[e] > v1) { v1 = p[e]; i1 = e; }
    gates[t * 2 + 0] = v0 * inv;
    gates[t * 2 + 1] = v1 * inv;
    eids[t * 2 + 0]  = i0;
    eids[t * 2 + 1]  = i1;
  }
}

// ---------------------------------------------------------------------------
// Kernel 2: zero d_out (float4 stores).
// ---------------------------------------------------------------------------
__global__ __launch_bounds__(256) void moe_zero_kernel(float4* __restrict__ p,
                                                       int n4) {
  const int i = blockIdx.x * blockDim.x + threadIdx.x;
  if (i < n4) p[i] = make_float4(0.f, 0.f, 0.f, 0.f);
}

// ---------------------------------------------------------------------------
// Kernel 3: build per-expert token lists. list entry encodes (token<<1)|k so
// the gate can be fetched as gates[code].
// ---------------------------------------------------------------------------
__global__ __launch_bounds__(256) void moe_assign_kernel(
    const int* __restrict__ eids, int* __restrict__ cnt,
    int* __restrict__ list) {
  const int t = blockIdx.x * blockDim.x + threadIdx.x;
  if (t >= NT_) return;
#pragma unroll
  for (int k = 0; k < 2; ++k) {
    const int e = eids[t * 2 + k];
    const int slot = atomicAdd(&cnt[e], 1);
    list[e * MAXA_ + slot] = (t << 1) | k;
  }
}

// ---------------------------------------------------------------------------
// Kernel 4: fused expert FFN. Block = (expert e, tile of 32 tokens), 8 waves.
//   xs (LDS)  : 32 x 768 gathered token rows
//   loop H in chunks of 128:
//     GEMM1: h = relu(xs @ W1_e[:,chunk] + b1)   (wmma f32 16x16x4)
//     GEMM2: y += h @ W2_e[chunk,:]              (wmma f32 16x16x4)
//   out[token] += gate * (y + b2)   via 2-way-commutative float atomics
// ---------------------------------------------------------------------------
__global__ __launch_bounds__(256) void moe_expert_kernel(
    const float* __restrict__ x,  const float* __restrict__ W1,
    const float* __restrict__ b1, const float* __restrict__ W2,
    const float* __restrict__ b2, const float* __restrict__ gates,
    const int* __restrict__ list, const int* __restrict__ cnt,
    float* __restrict__ out) {
  __shared__ float xs[TM_ * XSS_];
  __shared__ float hs[TM_ * HSS_];
  __shared__ float gate_s[TM_];
  __shared__ int   tok_s[TM_];

  const int e     = blockIdx.y;
  const int tile0 = blockIdx.x * TM_;
  const int n     = cnt[e];
  if (tile0 >= n) return;

  const int tid  = threadIdx.x;
  const int lane = tid & 31;
  const int w    = tid >> 5;     // wave id 0..7
  const int nn   = lane & 15;    // N / row index within 16
  const int hi   = lane >> 4;    // K-half select (0 -> K=0,1 ; 1 -> K=2,3)

  if (tid < TM_) {
    const int idx = tile0 + tid;
    if (idx < n) {
      const int code = list[e * MAXA_ + idx];
      tok_s[tid]  = code >> 1;
      gate_s[tid] = gates[code];
    } else {
      tok_s[tid]  = 0;           // safe address; result masked by gate==0
      gate_s[tid] = 0.0f;
    }
  }
  __syncthreads();

  // Gather 32 token rows of x into LDS (float4, conflict-free padded stride).
  for (int i = tid; i < TM_ * (C_ / 4); i += 256) {
    const int r  = i / (C_ / 4);
    const int c4 = i % (C_ / 4);
    const float4 xv =
        ((const float4*)(x + (size_t)tok_s[r] * C_))[c4];
    *(float4*)&xs[r * XSS_ + c4 * 4] = xv;
  }
  __syncthreads();

  const float* W1e = W1 + (size_t)e * C_ * H_;
  const float* W2e = W2 + (size_t)e * H_ * C_;

  const v8f zero8 = {0.f, 0.f, 0.f, 0.f, 0.f, 0.f, 0.f, 0.f};
  v8f acc[2][6];
#pragma unroll
  for (int j = 0; j < 6; ++j) { acc[0][j] = zero8; acc[1][j] = zero8; }

  for (int c0 = 0; c0 < H_; c0 += HC_) {
    // ---- GEMM1: this wave computes h cols [c0+w*16, +16), all 32 rows ----
    v8f h0 = zero8, h1 = zero8;
    const int  hcol = c0 + w * 16 + nn;
    const float* w1p = W1e + hcol;
    for (int k = 0; k < C_; k += 4) {
      const int kk = k + 2 * hi;
      v2f bm; bm.x = w1p[(size_t)kk * H_];
              bm.y = w1p[(size_t)(kk + 1) * H_];
      v2f a0; a0.x = xs[nn * XSS_ + kk];
              a0.y = xs[nn * XSS_ + kk + 1];
      v2f a1; a1.x = xs[(nn + 16) * XSS_ + kk];
              a1.y = xs[(nn + 16) * XSS_ + kk + 1];
      h0 = wmma_f32_k4(a0, bm, h0);
      h1 = wmma_f32_k4(a1, bm, h1);
    }
    const float bias = b1[e * H_ + c0 + w * 16 + nn];
#pragma unroll
    for (int v = 0; v < 8; ++v) {
      float t0 = h0[v] + bias; t0 = t0 > 0.f ? t0 : 0.f;
      float t1 = h1[v] + bias; t1 = t1 > 0.f ? t1 : 0.f;
      hs[(v + 8 * hi) * HSS_ + w * 16 + nn]        = t0;
      hs[(16 + v + 8 * hi) * HSS_ + w * 16 + nn]   = t1;
    }
    __syncthreads();

    // ---- GEMM2: y[2 row tiles][6 col tiles of this wave] += h @ W2 ----
    for (int k = 0; k < HC_; k += 4) {
      const int kk = k + 2 * hi;
      v2f a0; a0.x = hs[nn * HSS_ + kk];
              a0.y = hs[nn * HSS_ + kk + 1];
      v2f a1; a1.x = hs[(nn + 16) * HSS_ + kk];
              a1.y = hs[(nn + 16) * HSS_ + kk + 1];
      const float* w2row = W2e + (size_t)(c0 + kk) * C_ + nn;
#pragma unroll
      for (int j = 0; j < 6; ++j) {
        const float* w2p = w2row + (w * 6 + j) * 16;
        v2f bm; bm.x = w2p[0];
                bm.y = w2p[C_];
        acc[0][j] = wmma_f32_k4(a0, bm, acc[0][j]);
        acc[1][j] = wmma_f32_k4(a1, bm, acc[1][j]);
      }
    }
    __syncthreads();
  }

  // ---- epilogue: out[token] += gate * (y + b2) ----
#pragma unroll
  for (int j = 0; j < 6; ++j) {
    const int col = (w * 6 + j) * 16 + nn;
    const float b2v = b2[e * C_ + col];
#pragma unroll
    for (int v = 0; v < 8; ++v) {
      const int r0 = v + 8 * hi;
      const float g0 = gate_s[r0];
      if (g0 != 0.0f)
        atomicAdd(&out[(size_t)tok_s[r0] * C_ + col], g0 * (acc[0][j][v] + b2v));
      const int r1 = 16 + v + 8 * hi;
      const float g1 = gate_s[r1];
      if (g1 != 0.0f)
        atomicAdd(&out[(size_t)tok_s[r1] * C_ + col], g1 * (acc[1][j][v] + b2v));
    }
  }
}

// ---------------------------------------------------------------------------
extern "C" void kernel_launch(void* const* d_in, const int* in_sizes, int n_in,
                              void* d_out, int out_size, void* d_ws,
                              size_t ws_size, hipStream_t stream) {
  const float* x  = (const float*)d_in[0];
  const float* Wg = (const float*)d_in[1];
  const float* W1 = (const float*)d_in[2];
  const float* b1 = (const float*)d_in[3];
  const float* W2 = (const float*)d_in[4];
  const float* b2 = (const float*)d_in[5];
  float* out = (float*)d_out;

  // workspace layout
  char* ws = (char*)d_ws;
  int*   cnt   = (int*)ws;                          // 8 ints (pad 64B)
  float* gates = (float*)(ws + 64);                 // 8192 f32
  int*   eids  = (int*)(ws + 64 + 4 * 2 * NT_);     // 8192 i32
  int*   list  = (int*)(ws + 64 + 8 * 2 * NT_);     // 8 * 8192 i32

  // 1) router (also zeroes cnt): one wave per token
  moe_router_kernel<<<(NT_ * 32) / 256, 256, 0, stream>>>(x, Wg, gates, eids,
                                                          cnt);
  // 2) zero output
  const int n4 = out_size / 4;
  moe_zero_kernel<<<(n4 + 255) / 256, 256, 0, stream>>>((float4*)out, n4);
  // 3) build expert lists
  moe_assign_kernel<<<NT_ / 256, 256, 0, stream>>>(eids, cnt, list);
  // 4) fused expert FFN (grid: max tiles x experts; empty tiles exit early)
  dim3 grid(MAXA_ / TM_, E_);
  moe_expert_kernel<<<grid, 256, 0, stream>>>(x, W1, b1, W2, b2, gates, list,
                                              cnt, out);
}